// ParaGraphGNNLayer_7310034338072
// MI455X (gfx1250) — compile-verified
//
#include <hip/hip_runtime.h>
#include <stdint.h>

#define N_NODES   100000
#define N_EDGES   1000000
#define D         128
#define OUT_D     128
#define NUM_TYPES 5
#define NTILES    (N_NODES / 16)   // 6250 exact

typedef float v2f __attribute__((ext_vector_type(2)));
typedef float v8f __attribute__((ext_vector_type(8)));

// ---------------------------------------------------------------------------
// Kernel A: nfl = nh @ W_nf^T  via V_WMMA_F32_16X16X4_F32.
// One wave per 16-row node tile; 8 N-tiles (128 cols) accumulated in regs.
// ---------------------------------------------------------------------------
__global__ __launch_bounds__(256) void k_node_gemm(const float* __restrict__ nh,
                                                   const float* __restrict__ Wnf,
                                                   float* __restrict__ nfl) {
  const int lane = threadIdx.x & 31;
  const int wave = threadIdx.x >> 5;
  const int tile = blockIdx.x * 8 + wave;
  if (tile >= NTILES) return;                 // whole-wave exit: EXEC stays all-1s
  const int row0 = tile * 16;
  const int lsel = lane >> 4;                 // 0: K pair (k,k+1), 1: (k+2,k+3)
  const int lrow = lane & 15;

  v8f c[8] = {};
  const float* aBase = nh + (size_t)(row0 + lrow) * D;
#pragma unroll 4
  for (int ks = 0; ks < D / 4; ++ks) {
    const int col = ks * 4 + 2 * lsel;
    v2f a = *(const v2f*)(aBase + col);
#pragma unroll
    for (int nt = 0; nt < 8; ++nt) {
      v2f b = *(const v2f*)(Wnf + (size_t)(nt * 16 + lrow) * D + col); // B[k][n]=Wnf[n][k]
      c[nt] = __builtin_amdgcn_wmma_f32_16x16x4_f32(false, a, false, b,
                                                    (short)0, c[nt], false, false);
    }
  }
#pragma unroll
  for (int nt = 0; nt < 8; ++nt) {
#pragma unroll
    for (int v = 0; v < 8; ++v) {
      const int r = row0 + v + 8 * lsel;
      nfl[(size_t)r * D + nt * 16 + lrow] = c[nt][v];
    }
  }
}

// ---------------------------------------------------------------------------
// Kernel B: per-node attention scalars s = nfl@a_src, t = nfl@a_dst.
// One wave per node, float4 per lane, wave32 shuffle reduction.
// ---------------------------------------------------------------------------
__global__ __launch_bounds__(256) void k_proj_st(const float* __restrict__ nfl,
                                                 const float* __restrict__ Wattn,
                                                 float* __restrict__ s,
                                                 float* __restrict__ t) {
  const int lane = threadIdx.x & 31;
  const int wave = threadIdx.x >> 5;
  const int node = blockIdx.x * 8 + wave;
  if (node >= N_NODES) return;
  const float4 v  = *(const float4*)(nfl + (size_t)node * D + lane * 4);
  const float4 as = *(const float4*)(Wattn + lane * 4);
  const float4 ad = *(const float4*)(Wattn + D + lane * 4);
  float ps = v.x * as.x + v.y * as.y + v.z * as.z + v.w * as.w;
  float pt = v.x * ad.x + v.y * ad.y + v.z * ad.z + v.w * ad.w;
#pragma unroll
  for (int off = 16; off > 0; off >>= 1) {
    ps += __shfl_xor(ps, off, 32);
    pt += __shfl_xor(pt, off, 32);
  }
  if (lane == 0) { s[node] = ps; t[node] = pt; }
}

// ---------------------------------------------------------------------------
// Kernel C: ef = leaky(s[src]+t[dst]); seg = dst*5+y; atomic segment max
// via monotonic unsigned encoding (init key 0 == -inf).
// ---------------------------------------------------------------------------
__global__ __launch_bounds__(256) void k_edge_score(const long long* __restrict__ ei,
                                                    const long long* __restrict__ ey,
                                                    const float* __restrict__ s,
                                                    const float* __restrict__ t,
                                                    float* __restrict__ ef_buf,
                                                    int* __restrict__ seg_buf,
                                                    unsigned* __restrict__ m) {
  const int e = blockIdx.x * blockDim.x + threadIdx.x;
  if (e >= N_EDGES) return;
  const int src = (int)ei[e];
  const int dst = (int)ei[N_EDGES + e];
  float ef = s[src] + t[dst];
  ef = ef >= 0.f ? ef : 0.2f * ef;
  const int seg = dst * NUM_TYPES + (int)ey[e];
  ef_buf[e] = ef;
  seg_buf[e] = seg;
  const unsigned b = __float_as_uint(ef);
  const unsigned key = (b & 0x80000000u) ? ~b : (b | 0x80000000u);
  atomicMax(m + seg, key);
}

// ---------------------------------------------------------------------------
// Kernel D: ex = exp(ef - max[seg]); denom[seg] += ex  (ex overwrites ef buf)
// ---------------------------------------------------------------------------
__global__ __launch_bounds__(256) void k_edge_exp(float* __restrict__ efex_buf,
                                                  const int* __restrict__ seg_buf,
                                                  const unsigned* __restrict__ m,
                                                  float* __restrict__ denom) {
  const int e = blockIdx.x * blockDim.x + threadIdx.x;
  if (e >= N_EDGES) return;
  const int seg = seg_buf[e];
  const unsigned key = m[seg];
  const unsigned b = (key & 0x80000000u) ? (key & 0x7FFFFFFFu) : ~key;
  const float ex = expf(efex_buf[e] - __uint_as_float(b));
  efex_buf[e] = ex;
  atomicAdd(denom + seg, ex);
}

// ---------------------------------------------------------------------------
// Kernel E: h[dst] += (ex/denom[seg]) * nfl[src].  One wave per edge,
// float4 gather per lane + 4 f32 atomic adds (L2-resident).
// ---------------------------------------------------------------------------
__global__ __launch_bounds__(256) void k_aggregate(const long long* __restrict__ ei,
                                                   const int* __restrict__ seg_buf,
                                                   const float* __restrict__ ex_buf,
                                                   const float* __restrict__ denom,
                                                   const float* __restrict__ nfl,
                                                   float* __restrict__ h) {
  const int lane = threadIdx.x & 31;
  const int wave = threadIdx.x >> 5;
  const int e = blockIdx.x * 8 + wave;
  if (e >= N_EDGES) return;
  const int src = (int)ei[e];
  const int dst = (int)ei[N_EDGES + e];
  const float alpha = ex_buf[e] / denom[seg_buf[e]];
  const float4 v = *(const float4*)(nfl + (size_t)src * D + lane * 4);
  float* hp = h + (size_t)dst * D + lane * 4;
  atomicAdd(hp + 0, alpha * v.x);
  atomicAdd(hp + 1, alpha * v.y);
  atomicAdd(hp + 2, alpha * v.z);
  atomicAdd(hp + 3, alpha * v.w);
}

// ---------------------------------------------------------------------------
// Kernel F: out = relu(concat(nh,h) @ W_out^T + b).  K=256 fp32 WMMA.
// ---------------------------------------------------------------------------
__global__ __launch_bounds__(256) void k_out_gemm(const float* __restrict__ nh,
                                                  const float* __restrict__ h,
                                                  const float* __restrict__ Wout,
                                                  const float* __restrict__ bout,
                                                  float* __restrict__ out) {
  const int lane = threadIdx.x & 31;
  const int wave = threadIdx.x >> 5;
  const int tile = blockIdx.x * 8 + wave;
  if (tile >= NTILES) return;
  const int row0 = tile * 16;
  const int lsel = lane >> 4;
  const int lrow = lane & 15;

  v8f c[8] = {};
  const float* aN = nh + (size_t)(row0 + lrow) * D;
  const float* aH = h  + (size_t)(row0 + lrow) * D - D;   // indexed with col in [128,256)
#pragma unroll 2
  for (int ks = 0; ks < (2 * D) / 4; ++ks) {
    const int col = ks * 4 + 2 * lsel;
    const float* aBase = (ks < 32) ? aN : aH;             // uniform per ks
    v2f a = *(const v2f*)(aBase + col);
#pragma unroll
    for (int nt = 0; nt < 8; ++nt) {
      v2f b = *(const v2f*)(Wout + (size_t)(nt * 16 + lrow) * (2 * D) + col);
      c[nt] = __builtin_amdgcn_wmma_f32_16x16x4_f32(false, a, false, b,
                                                    (short)0, c[nt], false, false);
    }
  }
#pragma unroll
  for (int nt = 0; nt < 8; ++nt) {
#pragma unroll
    for (int v = 0; v < 8; ++v) {
      const int r = row0 + v + 8 * lsel;
      const int ccol = nt * 16 + lrow;
      float val = c[nt][v] + bout[ccol];
      out[(size_t)r * OUT_D + ccol] = val > 0.f ? val : 0.f;
    }
  }
}

// ---------------------------------------------------------------------------
extern "C" void kernel_launch(void* const* d_in, const int* in_sizes, int n_in,
                              void* d_out, int out_size, void* d_ws, size_t ws_size,
                              hipStream_t stream) {
  const float*     nh    = (const float*)d_in[0];
  const long long* ey    = (const long long*)d_in[1];
  const long long* ei    = (const long long*)d_in[2];
  const float*     Wnf   = (const float*)d_in[3];
  const float*     Wattn = (const float*)d_in[4];
  const float*     Wout  = (const float*)d_in[5];
  const float*     bout  = (const float*)d_in[6];
  float*           out   = (float*)d_out;

  char* ws = (char*)d_ws;
  size_t off = 0;
  auto carve = [&](size_t bytes) -> char* {
    char* p = ws + off;
    off = (off + bytes + 255) & ~(size_t)255;
    return p;
  };
  float*    nfl   = (float*)carve((size_t)N_NODES * D * 4);           // 51.2 MB
  float*    s     = (float*)carve((size_t)N_NODES * 4);
  float*    t     = (float*)carve((size_t)N_NODES * 4);
  unsigned* m     = (unsigned*)carve((size_t)N_NODES * NUM_TYPES * 4); // 2 MB
  float*    denom = (float*)carve((size_t)N_NODES * NUM_TYPES * 4);    // 2 MB
  float*    efex  = (float*)carve((size_t)N_EDGES * 4);                // 4 MB
  int*      segb  = (int*)carve((size_t)N_EDGES * 4);                  // 4 MB
  float*    h     = (float*)carve((size_t)N_NODES * D * 4);            // 51.2 MB

  hipMemsetAsync(m,     0, (size_t)N_NODES * NUM_TYPES * 4, stream);
  hipMemsetAsync(denom, 0, (size_t)N_NODES * NUM_TYPES * 4, stream);
  hipMemsetAsync(h,     0, (size_t)N_NODES * D * 4, stream);

  k_node_gemm <<<(NTILES + 7) / 8,        256, 0, stream>>>(nh, Wnf, nfl);
  k_proj_st   <<<(N_NODES + 7) / 8,       256, 0, stream>>>(nfl, Wattn, s, t);
  k_edge_score<<<(N_EDGES + 255) / 256,   256, 0, stream>>>(ei, ey, s, t, efex, segb, m);
  k_edge_exp  <<<(N_EDGES + 255) / 256,   256, 0, stream>>>(efex, segb, m, denom);
  k_aggregate <<<(N_EDGES + 7) / 8,       256, 0, stream>>>(ei, segb, efex, denom, nfl, h);
  k_out_gemm  <<<(NTILES + 7) / 8,        256, 0, stream>>>(nh, h, Wout, bout, out);
}